// SAGE_49778670961292
// MI455X (gfx1250) — compile-verified
//
#include <hip/hip_runtime.h>
#include <stdint.h>

#define N_NODES 100000
#define N_EDGES 1600000
#define DIN     128   // input feature dim of every layer's aggregation

typedef __attribute__((ext_vector_type(2))) float v2f;
typedef __attribute__((ext_vector_type(8))) float v8f;

// ---------------------------------------------------------------- utilities
__global__ void zero_f32(float* __restrict__ p, long long n) {
  long long i = (long long)blockIdx.x * blockDim.x + threadIdx.x;
  if (i < n) p[i] = 0.0f;
}

// deg[dst] += 1 over all edges (float so we can reuse atomicAdd f32 path)
__global__ void degree_kernel(const long long* __restrict__ dst,
                              float* __restrict__ deg) {
  int e = blockIdx.x * blockDim.x + threadIdx.x;
  if (e < N_EDGES) atomicAdd(&deg[(int)dst[e]], 1.0f);
}

// deg -> 1 / max(deg, 1)
__global__ void rdeg_kernel(float* __restrict__ deg) {
  int i = blockIdx.x * blockDim.x + threadIdx.x;
  if (i < N_NODES) deg[i] = 1.0f / fmaxf(deg[i], 1.0f);
}

// msg[dst, :] += x[src, :]   one thread = one edge x 4 consecutive features
// (msg fits in 192MB L2 -> atomics resolve in L2)
__global__ void scatter_kernel(const long long* __restrict__ src,
                               const long long* __restrict__ dst,
                               const float* __restrict__ x,
                               float* __restrict__ msg) {
  long long t = (long long)blockIdx.x * blockDim.x + threadIdx.x;
  const int chunks = DIN / 4;                 // 32
  if (t >= (long long)N_EDGES * chunks) return;
  int e = (int)(t >> 5);                      // t / 32
  int c = ((int)t & 31) * 4;
  int s = (int)src[e];
  int d = (int)dst[e];
  const float* xs = x   + (long long)s * DIN + c;
  float*       mp = msg + (long long)d * DIN + c;
  atomicAdd(mp + 0, xs[0]);
  atomicAdd(mp + 1, xs[1]);
  atomicAdd(mp + 2, xs[2]);
  atomicAdd(mp + 3, xs[3]);
}

// Fused SAGE layer GEMM:
//   out[m,n] = act( sum_k (msg[m,k]*rdeg[m]) * Wl[k,n] + xin[m,k] * Wr[k,n] + b[n] )
// One wave = one 16(M) x 16(N) tile, fp32 WMMA 16x16x4, K looped in steps of 4.
// blockDim.x = 32 * (dout/16); grid.x = N_NODES/16.
__global__ __launch_bounds__(256)
void sage_gemm_kernel(const float* __restrict__ msg,
                      const float* __restrict__ rdeg,
                      const float* __restrict__ xin,
                      const float* __restrict__ Wl,
                      const float* __restrict__ Wr,
                      const float* __restrict__ bias,
                      float* __restrict__ out,
                      int dout, int relu) {
  const int lane  = threadIdx.x & 31;
  const int wave  = threadIdx.x >> 5;       // which 16-col tile
  const int row0  = blockIdx.x * 16;
  const int n0    = wave * 16;
  const int m     = lane & 15;              // A row within tile
  const int koff  = (lane >> 4) * 2;        // lane-half selects K pair (0 or 2)
  const int nlane = lane & 15;              // B/C column within tile

  const float  r      = rdeg[row0 + m];
  const float* msgRow = msg + (long long)(row0 + m) * DIN;
  const float* xRow   = xin + (long long)(row0 + m) * DIN;

  v8f acc = {};
  for (int k = 0; k < DIN; k += 4) {
    v2f a, b;
    // ---- neighbor-mean branch: A = mean tile, B = Wl tile
    a.x = msgRow[k + koff]     * r;
    a.y = msgRow[k + koff + 1] * r;
    b.x = Wl[(long long)(k + koff)     * dout + n0 + nlane];
    b.y = Wl[(long long)(k + koff + 1) * dout + n0 + nlane];
    acc = __builtin_amdgcn_wmma_f32_16x16x4_f32(
        /*neg_a=*/false, a, /*neg_b=*/false, b,
        /*c_mod=*/(short)0, acc, /*reuse_a=*/false, /*reuse_b=*/false);
    // ---- root branch: A = x tile, B = Wr tile
    a.x = xRow[k + koff];
    a.y = xRow[k + koff + 1];
    b.x = Wr[(long long)(k + koff)     * dout + n0 + nlane];
    b.y = Wr[(long long)(k + koff + 1) * dout + n0 + nlane];
    acc = __builtin_amdgcn_wmma_f32_16x16x4_f32(
        false, a, false, b, (short)0, acc, false, false);
  }

  // C/D layout: VGPR v -> row (row0 + v + (lane>=16 ? 8:0)), col n0 + (lane&15)
  const int rowBase = row0 + ((lane >> 4) ? 8 : 0);
  const float bn = bias[n0 + nlane];
#pragma unroll
  for (int v = 0; v < 8; ++v) {
    float val = acc[v] + bn;
    if (relu) val = fmaxf(val, 0.0f);
    out[(long long)(rowBase + v) * dout + n0 + nlane] = val;
  }
}

// ---------------------------------------------------------------- launcher
extern "C" void kernel_launch(void* const* d_in, const int* in_sizes, int n_in,
                              void* d_out, int out_size, void* d_ws, size_t ws_size,
                              hipStream_t stream) {
  // inputs: x, edge_index, Wl0, Wr0, b0, Wl1, Wr1, b1, Wl2, Wr2, b2
  const float*     x   = (const float*)d_in[0];
  const long long* ei  = (const long long*)d_in[1];   // (2, 1.6M) int64
  const float* Wl0 = (const float*)d_in[2];
  const float* Wr0 = (const float*)d_in[3];
  const float* b0  = (const float*)d_in[4];
  const float* Wl1 = (const float*)d_in[5];
  const float* Wr1 = (const float*)d_in[6];
  const float* b1  = (const float*)d_in[7];
  const float* Wl2 = (const float*)d_in[8];
  const float* Wr2 = (const float*)d_in[9];
  const float* b2  = (const float*)d_in[10];

  const long long* src = ei;
  const long long* dst = ei + N_EDGES;

  // workspace layout (floats)
  float* ws   = (float*)d_ws;
  float* deg  = ws;                                   // N_NODES
  float* msg  = deg + ((N_NODES + 255) & ~255);       // N_NODES*128
  float* h1   = msg + (long long)N_NODES * DIN;       // N_NODES*128
  float* h2   = h1  + (long long)N_NODES * DIN;       // N_NODES*128

  const int ZB = 256;
  const long long msgN = (long long)N_NODES * DIN;
  const long long scatT = (long long)N_EDGES * (DIN / 4);

  // degrees (shared by all layers)
  zero_f32<<<(N_NODES + ZB - 1) / ZB, ZB, 0, stream>>>(deg, N_NODES);
  degree_kernel<<<(N_EDGES + ZB - 1) / ZB, ZB, 0, stream>>>(dst, deg);
  rdeg_kernel<<<(N_NODES + ZB - 1) / ZB, ZB, 0, stream>>>(deg);

  const unsigned scatBlocks = (unsigned)((scatT + ZB - 1) / ZB);
  const unsigned zeroBlocks = (unsigned)((msgN + ZB - 1) / ZB);
  const unsigned gemmBlocks = N_NODES / 16;           // 6250

  // ---- layer 0: x -> h1 (128->128, relu)
  zero_f32<<<zeroBlocks, ZB, 0, stream>>>(msg, msgN);
  scatter_kernel<<<scatBlocks, ZB, 0, stream>>>(src, dst, x, msg);
  sage_gemm_kernel<<<gemmBlocks, 32 * (128 / 16), 0, stream>>>(
      msg, deg, x, Wl0, Wr0, b0, h1, 128, 1);

  // ---- layer 1: h1 -> h2 (128->128, relu)
  zero_f32<<<zeroBlocks, ZB, 0, stream>>>(msg, msgN);
  scatter_kernel<<<scatBlocks, ZB, 0, stream>>>(src, dst, h1, msg);
  sage_gemm_kernel<<<gemmBlocks, 32 * (128 / 16), 0, stream>>>(
      msg, deg, h1, Wl1, Wr1, b1, h2, 128, 1);

  // ---- layer 2: h2 -> out (128->64, no relu)
  zero_f32<<<zeroBlocks, ZB, 0, stream>>>(msg, msgN);
  scatter_kernel<<<scatBlocks, ZB, 0, stream>>>(src, dst, h2, msg);
  sage_gemm_kernel<<<gemmBlocks, 32 * (64 / 16), 0, stream>>>(
      msg, deg, h2, Wl2, Wr2, b2, (float*)d_out, 64, 0);
}